// SGC_9371618640573
// MI455X (gfx1250) — compile-verified
//
#include <hip/hip_runtime.h>
#include <hip/hip_bf16.h>

// ---------------------------------------------------------------------------
// SGC: h <- D^-1/2 (A+I) D^-1/2 h  (x3), then out = h @ W + b
// N=100000 nodes, E=1200000 edges, dim 64, fp32 throughout.
// Propagation is L2/atomic bound (~2 GB over 3 rounds); the GEMM is tiny and
// memory bound, so fp32 WMMA (v_wmma_f32_16x16x4_f32) keeps exact numerics
// at zero perf cost.
// ---------------------------------------------------------------------------

typedef float v2f __attribute__((ext_vector_type(2)));
typedef float v8f __attribute__((ext_vector_type(8)));

// ---------------- degree / normalization ----------------

__global__ __launch_bounds__(256) void deg_init(float* __restrict__ deg, int n) {
    int i = blockIdx.x * 256 + threadIdx.x;
    if (i < n) deg[i] = 1.0f;  // self-loop contribution
}

__global__ __launch_bounds__(256) void deg_count(const long long* __restrict__ dst,
                                                 float* __restrict__ deg, int e) {
    int i = blockIdx.x * 256 + threadIdx.x;
    if (i < e) unsafeAtomicAdd(&deg[(int)dst[i]], 1.0f);
}

__global__ __launch_bounds__(256) void deg_finalize(float* __restrict__ dinv, int n) {
    int i = blockIdx.x * 256 + threadIdx.x;
    if (i < n) dinv[i] = rsqrtf(dinv[i]);  // deg >= 1 always (self-loops)
}

// ---------------- propagation ----------------

// Self-loop seed: h_out[i,:] = dinv[i]^2 * h_in[i,:]   (fully overwrites h_out)
__global__ __launch_bounds__(256) void prop_init(float* __restrict__ hout,
                                                 const float* __restrict__ hin,
                                                 const float* __restrict__ dinv, int n) {
    int t = blockIdx.x * 256 + threadIdx.x;  // one thread per float4 (16 per row)
    int row = t >> 4;
    int q   = t & 15;
    if (row >= n) return;
    float w = dinv[row];
    w *= w;
    const float4 v = ((const float4*)(hin + (size_t)row * 64))[q];
    float4 o = make_float4(v.x * w, v.y * w, v.z * w, v.w * w);
    ((float4*)(hout + (size_t)row * 64))[q] = o;
}

// Edge scatter: h_out[d,:] += dinv[s]*dinv[d] * h_in[s,:]
__global__ __launch_bounds__(256) void prop_edges(float* __restrict__ hout,
                                                  const float* __restrict__ hin,
                                                  const float* __restrict__ dinv,
                                                  const long long* __restrict__ src,
                                                  const long long* __restrict__ dst, int e) {
    int t   = blockIdx.x * 256 + threadIdx.x;  // 16 threads per edge
    int eid = t >> 4;
    int q   = t & 15;
    if (eid >= e) return;
    int s = (int)src[eid];
    int d = (int)dst[eid];
    float w = dinv[s] * dinv[d];
    const float4 v = ((const float4*)(hin + (size_t)s * 64))[q];
    float* o = hout + (size_t)d * 64 + q * 4;
    unsafeAtomicAdd(o + 0, w * v.x);
    unsafeAtomicAdd(o + 1, w * v.y);
    unsafeAtomicAdd(o + 2, w * v.z);
    unsafeAtomicAdd(o + 3, w * v.w);
}

// ---------------- final GEMM: out = h @ W + b, via V_WMMA_F32_16X16X4_F32 ----
//
// One wave computes a 16-row x 64-col slab: four 16x16 C tiles, K=64 in steps
// of 4 -> 64 wmma ops per wave.
//
// W is staged into LDS in *fragment order*: Wp[k][c] = {W[2k][c], W[2k+1][c]}
// (float2, 32x64 = 16 KB) so each lane's B fragment is one contiguous
// ds_load_b64 straight into an aligned VGPR pair -- no repack movs.
//
// A (16x4 f32) VGPR layout: lanes 0-15 hold K={k0,k0+1}, lanes 16-31 K={k0+2,k0+3}
// B (4x16  f32): VGPR0 = row (k0 + khi), VGPR1 = row (k0 + khi + 1), N = lane%16
// C/D (16x16 f32): VGPR v: lanes 0-15 -> row v, lanes 16-31 -> row v+8, N = lane%16

__global__ __launch_bounds__(256) void sgc_gemm_wmma(const float* __restrict__ h,
                                                     const float* __restrict__ Wm,
                                                     const float* __restrict__ bias,
                                                     float* __restrict__ out, int nrows) {
    __shared__ v2f Wp[32 * 64];  // Wp[k][c] = {W[2k][c], W[2k+1][c]}, 16 KB
    // cooperative pack: 2048 float2 elements by 256 threads (8 iters)
    for (int i = threadIdx.x; i < 32 * 64; i += 256) {
        const int k = i >> 6;   // row pair 0..31
        const int c = i & 63;   // col
        v2f p;
        p.x = Wm[(2 * k + 0) * 64 + c];
        p.y = Wm[(2 * k + 1) * 64 + c];
        Wp[i] = p;
    }
    __syncthreads();

    const int wave = threadIdx.x >> 5;  // 0..7
    const int lane = threadIdx.x & 31;
    const int row_base = (blockIdx.x * 8 + wave) * 16;
    if (row_base >= nrows) return;  // nrows % 16 == 0: whole wave active or idle

    const int lrow = lane & 15;
    const int half = lane >> 4;      // 0 for lanes 0-15, 1 for lanes 16-31
    const int khi  = half * 2;

    v8f acc[4] = {v8f{}, v8f{}, v8f{}, v8f{}};

    const float* arow = h + (size_t)(row_base + lrow) * 64;
#pragma unroll 4
    for (int k0 = 0; k0 < 64; k0 += 4) {
        // A fragment: contiguous float2 at K = k0 + khi
        v2f afrag = *(const v2f*)(arow + k0 + khi);
        const v2f* brow = Wp + ((k0 >> 1) + half) * 64;  // pair index
#pragma unroll
        for (int ct = 0; ct < 4; ++ct) {
            v2f bfrag = brow[ct * 16 + lrow];  // one ds_load_b64
            acc[ct] = __builtin_amdgcn_wmma_f32_16x16x4_f32(
                /*neg_a=*/false, afrag, /*neg_b=*/false, bfrag,
                /*c_mod=*/(short)0, acc[ct], /*reuse_a=*/false, /*reuse_b=*/false);
        }
    }

    const int rofs = half * 8;  // lanes 16-31 hold rows v+8
#pragma unroll
    for (int ct = 0; ct < 4; ++ct) {
        const int col = ct * 16 + lrow;
        const float bb = bias[col];
#pragma unroll
        for (int v = 0; v < 8; ++v)
            out[(size_t)(row_base + rofs + v) * 64 + col] = acc[ct][v] + bb;
    }
}

// ---------------- launch ----------------

extern "C" void kernel_launch(void* const* d_in, const int* in_sizes, int n_in,
                              void* d_out, int out_size, void* d_ws, size_t ws_size,
                              hipStream_t stream) {
    const float*     x    = (const float*)d_in[0];
    const long long* ei   = (const long long*)d_in[1];  // int64 edge_index [2,E]
    const float*     Wm   = (const float*)d_in[2];
    const float*     bias = (const float*)d_in[3];
    float*           out  = (float*)d_out;

    const int N = in_sizes[0] / 64;
    const int E = in_sizes[1] / 2;
    const long long* src = ei;
    const long long* dst = ei + E;

    // scratch layout: dinv[N] | hA[N*64]   (~26 MB)
    float* dinv = (float*)d_ws;
    float* hA   = dinv + (((size_t)N + 3) & ~(size_t)3);

    const int nb_n  = (N + 255) / 256;
    const int nb_e  = (E + 255) / 256;
    const int nb_n4 = (N * 16 + 255) / 256;  // one thread per float4
    const int nb_e4 = (E * 16 + 255) / 256;  // 16 threads per edge

    // normalization
    deg_init<<<nb_n, 256, 0, stream>>>(dinv, N);
    deg_count<<<nb_e, 256, 0, stream>>>(dst, dinv, E);
    deg_finalize<<<nb_n, 256, 0, stream>>>(dinv, N);

    // round 1: hA = S x
    prop_init<<<nb_n4, 256, 0, stream>>>(hA, x, dinv, N);
    prop_edges<<<nb_e4, 256, 0, stream>>>(hA, x, dinv, src, dst, E);
    // round 2: out = S hA   (d_out used as ping-pong buffer)
    prop_init<<<nb_n4, 256, 0, stream>>>(out, hA, dinv, N);
    prop_edges<<<nb_e4, 256, 0, stream>>>(out, hA, dinv, src, dst, E);
    // round 3: hA = S out
    prop_init<<<nb_n4, 256, 0, stream>>>(hA, out, dinv, N);
    prop_edges<<<nb_e4, 256, 0, stream>>>(hA, out, dinv, src, dst, E);

    // out = hA @ W + b  (fp32 WMMA)
    const int waves = N / 16;                 // N % 16 == 0 for N=100000
    const int nb_g  = (waves + 7) / 8;
    sgc_gemm_wmma<<<nb_g, 256, 0, stream>>>(hA, Wm, bias, out, N);
}